// GRU_GAT_RRGPredictor_14328010900024
// MI455X (gfx1250) — compile-verified
//
#include <hip/hip_runtime.h>

// ---------------------------------------------------------------------------
// CDNA5 / gfx1250, wave32.  All matmuls use V_WMMA_F32_16X16X4_F32 (fp32 in,
// fp32 acc): the problem is memory-bound (~8.6 GFLOP vs ~100MB traffic at
// 23.3 TB/s => ~4us floor), so fp32 WMMA preserves reference precision with
// throughput to spare.  LDS row stride padded to 260 floats so the attention
// GEMM's ds_load_b64 A-fragment reads are bank-conflict-free (64 banks x 4B:
// bank = (4r + k) mod 64, distinct per row r).
// ---------------------------------------------------------------------------

typedef float v2f __attribute__((ext_vector_type(2)));
typedef float v8f __attribute__((ext_vector_type(8)));

#define WMMA_F32(A, Bv, C) \
  __builtin_amdgcn_wmma_f32_16x16x4_f32(false, (A), false, (Bv), (short)0, (C), false, false)

constexpr int Bb  = 64;          // batch
constexpr int Nn  = 256;         // nodes
constexpr int INF = 128;         // in features
constexpr int OF  = 128;         // out features per head
constexpr int Hh  = 4;           // heads
constexpr int MF  = Bb * Nn;     // 16384 rows
constexpr int KC  = OF * Hh;     // 512 concat width
constexpr int PSTR = 260;        // padded LDS row stride (floats), even, !=0 mod 64 pattern
constexpr float NEG_SLOPE = 0.2f;

// ---------------------------------------------------------------------------
// k1: Wh = h @ W   (16384x128)*(128x512).  One wave per 16x16 output tile.
// ---------------------------------------------------------------------------
__global__ __launch_bounds__(256) void k1_wh(const float* __restrict__ h,
                                             const float* __restrict__ W,
                                             float* __restrict__ Wh) {
  const int wave = threadIdx.x >> 5, lane = threadIdx.x & 31;
  const int tile = blockIdx.x * 8 + wave;     // 32768 tiles = 1024 x 32
  const int mt = tile >> 5, nt = tile & 31;
  const int row0 = mt * 16, col0 = nt * 16;
  const int r = lane & 15, kb = (lane >> 4) << 1;

  v8f acc = {};
  const float* arow = h + (size_t)(row0 + r) * INF;
#pragma unroll
  for (int k = 0; k < INF; k += 4) {
    v2f a = *(const v2f*)(arow + k + kb);                 // A[l%16][k+kb .. +1]
    v2f b;
    b.x = W[(size_t)(k + kb + 0) * KC + col0 + r];        // B[k+kb+v][l%16]
    b.y = W[(size_t)(k + kb + 1) * KC + col0 + r];
    acc = WMMA_F32(a, b, acc);
  }
#pragma unroll
  for (int v = 0; v < 8; ++v) {
    const int row = row0 + v + ((lane >> 4) << 3);        // D[v+8*(l/16)][l%16]
    Wh[(size_t)row * KC + col0 + r] = acc[v];
  }
}

// ---------------------------------------------------------------------------
// k2: per-(b,h,n) scalars: e_i, e_j, s_proj, s_sim.  One wave per row (128-dot),
// float4 loads + __shfl_xor tree reduction (wave32).
// ---------------------------------------------------------------------------
__global__ __launch_bounds__(256) void k2_scalars(const float* __restrict__ Wh,
                                                  const float* __restrict__ a,
                                                  const float* __restrict__ wp,
                                                  const float* __restrict__ wsm,
                                                  float* __restrict__ ei,
                                                  float* __restrict__ ej,
                                                  float* __restrict__ sp,
                                                  float* __restrict__ ss) {
  const int wave = threadIdx.x >> 5, lane = threadIdx.x & 31;
  const int idx = blockIdx.x * 8 + wave;      // 65536 = B*H*N  (b,h,n order)
  const int b  = idx >> 10;
  const int hh = (idx >> 8) & 3;
  const int n  = idx & 255;

  const float* row = Wh + ((size_t)(b * Nn + n) * KC + hh * OF);
  const float* a1  = a + hh * (2 * OF);
  const float* a2  = a1 + OF;
  const int f = lane * 4;

  float4 w  = *(const float4*)(row + f);
  float4 v1 = *(const float4*)(a1  + f);
  float4 v2 = *(const float4*)(a2  + f);
  float4 vp = *(const float4*)(wp  + f);
  float4 vs = *(const float4*)(wsm + f);

  float s1 = w.x * v1.x + w.y * v1.y + w.z * v1.z + w.w * v1.w;
  float s2 = w.x * v2.x + w.y * v2.y + w.z * v2.z + w.w * v2.w;
  float s3 = w.x * vp.x + w.y * vp.y + w.z * vp.z + w.w * vp.w;
  float s4 = w.x * vs.x + w.y * vs.y + w.z * vs.z + w.w * vs.w;
#pragma unroll
  for (int off = 16; off; off >>= 1) {
    s1 += __shfl_xor(s1, off, 32);
    s2 += __shfl_xor(s2, off, 32);
    s3 += __shfl_xor(s3, off, 32);
    s4 += __shfl_xor(s4, off, 32);
  }
  if (lane == 0) { ei[idx] = s1; ej[idx] = s2; sp[idx] = s3; ss[idx] = s4; }
}

// ---------------------------------------------------------------------------
// k3: attention.  One wave per (b, h, 16-row tile).
//  Phase 1: logits + softmax (wave shuffle max/sum), exp(e-m) -> LDS (padded).
//  Phase 2: (16x256)@(256x128) with WMMA; A-frags via conflict-free
//           ds_load_b64 from LDS; rows scaled by 1/rowsum in the epilogue.
// ---------------------------------------------------------------------------
__global__ __launch_bounds__(32) void k3_attn(const float* __restrict__ Wh,
                                              const float* __restrict__ P,
                                              const float* __restrict__ S,
                                              const int*   __restrict__ Q,
                                              const float* __restrict__ qb,
                                              const float* __restrict__ ei,
                                              const float* __restrict__ ej,
                                              const float* __restrict__ sp,
                                              const float* __restrict__ ss,
                                              float* __restrict__ hp) {
  __shared__ float pbuf[16 * PSTR];  // exp(e - rowmax), padded rows (16.6KB)
  __shared__ float sinv[16];         // 1 / rowsum

  const int lane = threadIdx.x & 31;
  const int bid  = blockIdx.x;                 // 4096 = 64 * 4 * 16
  const int b    = bid >> 6;
  const int hh   = (bid >> 4) & 3;
  const int it   = bid & 15;
  const int i0   = it * 16;
  const int sbase = (b * Hh + hh) * Nn;

  // per-lane column slice: j = lane + 32*jj
  float ejv[8];
#pragma unroll
  for (int jj = 0; jj < 8; ++jj) ejv[jj] = ej[sbase + lane + jj * 32];

  for (int i = 0; i < 16; ++i) {
    const int gi = i0 + i;
    const float ei_r = ei[sbase + gi];
    const float sp_r = sp[sbase + gi];
    const float ss_r = ss[sbase + gi];
    const float* Prow = P + (size_t)(b * Nn + gi) * Nn;
    const float* Srow = S + (size_t)(b * Nn + gi) * Nn;
    const int*   Qrow = Q + (size_t)(b * Nn + gi) * Nn * 2;

    float ev[8];
    float m = -3.4e38f;
#pragma unroll
    for (int jj = 0; jj < 8; ++jj) {
      const int j = lane + jj * 32;
      float e = ei_r + ejv[jj];
      e = e > 0.f ? e : NEG_SLOPE * e;                 // leaky relu on (ei+ej)
      e += sp_r * Prow[j] + ss_r * Srow[j];
      const int q0 = Qrow[2 * j], q1 = Qrow[2 * j + 1];
      e += qb[q0 * 4 + q1];
      ev[jj] = e;
      m = fmaxf(m, e);
    }
#pragma unroll
    for (int off = 16; off; off >>= 1) m = fmaxf(m, __shfl_xor(m, off, 32));
    float sum = 0.f;
#pragma unroll
    for (int jj = 0; jj < 8; ++jj) {
      const float p = __expf(ev[jj] - m);
      sum += p;
      pbuf[i * PSTR + lane + jj * 32] = p;
    }
#pragma unroll
    for (int off = 16; off; off >>= 1) sum += __shfl_xor(sum, off, 32);
    if (lane == 0) sinv[i] = 1.0f / sum;
  }
  __syncthreads();   // single-wave WG: orders LDS writes before fragment reads

  // GEMM: A = pbuf (16x256, padded), B = Wh head block (256x128)
  const int r  = lane & 15;
  const int hl = lane >> 4;
  const int kb = hl << 1;
  const float* Whb = Wh + (size_t)(b * Nn) * KC + hh * OF;

  v8f acc[8];
#pragma unroll
  for (int t = 0; t < 8; ++t) acc[t] = (v8f){};

  for (int k = 0; k < Nn; k += 4) {
    v2f a = *(const v2f*)(&pbuf[r * PSTR + k + kb]);      // conflict-free b64
#pragma unroll
    for (int t = 0; t < 8; ++t) {
      v2f bv;
      bv.x = Whb[(size_t)(k + kb + 0) * KC + t * 16 + r];
      bv.y = Whb[(size_t)(k + kb + 1) * KC + t * 16 + r];
      acc[t] = WMMA_F32(a, bv, acc[t]);
    }
  }

#pragma unroll
  for (int v = 0; v < 8; ++v) {
    const int ri  = v + (hl << 3);
    const float sc = sinv[ri];
    const size_t grow = (size_t)(b * Nn + i0 + ri) * KC + hh * OF;
#pragma unroll
    for (int t = 0; t < 8; ++t)
      hp[grow + t * 16 + r] = acc[t][v] * sc;
  }
}

// ---------------------------------------------------------------------------
// k4: out = elu( h_prime(16384x512) @ W_concat(512x128) )
// ---------------------------------------------------------------------------
__global__ __launch_bounds__(256) void k4_out(const float* __restrict__ hp,
                                              const float* __restrict__ Wc,
                                              float* __restrict__ out) {
  const int wave = threadIdx.x >> 5, lane = threadIdx.x & 31;
  const int tile = blockIdx.x * 8 + wave;     // 8192 tiles = 1024 x 8
  const int mt = tile >> 3, nt = tile & 7;
  const int row0 = mt * 16, col0 = nt * 16;
  const int r = lane & 15, kb = (lane >> 4) << 1;

  v8f acc = {};
  const float* arow = hp + (size_t)(row0 + r) * KC;
  for (int k = 0; k < KC; k += 4) {
    v2f a = *(const v2f*)(arow + k + kb);
    v2f b;
    b.x = Wc[(size_t)(k + kb + 0) * OF + col0 + r];
    b.y = Wc[(size_t)(k + kb + 1) * OF + col0 + r];
    acc = WMMA_F32(a, b, acc);
  }
#pragma unroll
  for (int v = 0; v < 8; ++v) {
    float x = acc[v];
    x = x > 0.f ? x : (__expf(x) - 1.0f);               // ELU (alpha = 1)
    const int row = row0 + v + ((lane >> 4) << 3);
    out[(size_t)row * OF + col0 + r] = x;
  }
}

// ---------------------------------------------------------------------------
extern "C" void kernel_launch(void* const* d_in, const int* in_sizes, int n_in,
                              void* d_out, int out_size, void* d_ws, size_t ws_size,
                              hipStream_t stream) {
  const float* h   = (const float*)d_in[0];   // (B,N,IN_F)
  const float* P   = (const float*)d_in[1];   // (B,N,N)
  const float* S   = (const float*)d_in[2];   // (B,N,N)
  const int*   Q   = (const int*)  d_in[3];   // (B,N,N,2)
  const float* W   = (const float*)d_in[4];   // (IN_F, OF*H)
  const float* a   = (const float*)d_in[5];   // (H, 2*OF, 1)
  const float* wp  = (const float*)d_in[6];   // (OF,)
  const float* wsm = (const float*)d_in[7];   // (OF,)
  const float* qb  = (const float*)d_in[8];   // (4,4)
  const float* Wc  = (const float*)d_in[9];   // (OF*H, OF)
  float* out = (float*)d_out;                 // (B,N,OF)

  // workspace layout (~65 MB)
  float* Wh  = (float*)d_ws;                  // 16384 x 512
  float* hp  = Wh  + (size_t)MF * KC;         // 16384 x 512
  float* ei  = hp  + (size_t)MF * KC;         // B*H*N
  float* ejp = ei  + Bb * Hh * Nn;
  float* spp = ejp + Bb * Hh * Nn;
  float* ssp = spp + Bb * Hh * Nn;

  k1_wh     <<<4096, 256, 0, stream>>>(h, W, Wh);
  k2_scalars<<<8192, 256, 0, stream>>>(Wh, a, wp, wsm, ei, ejp, spp, ssp);
  k3_attn   <<<4096,  32, 0, stream>>>(Wh, P, S, Q, qb, ei, ejp, spp, ssp, hp);
  k4_out    <<<1024, 256, 0, stream>>>(hp, Wc, out);
}